// GCN_44212393345339
// MI455X (gfx1250) — compile-verified
//
#include <hip/hip_runtime.h>
#include <hip/hip_bf16.h>

// ---------------------------------------------------------------------------
// RGCN (2 layers, mean-per-(dst,rel) aggregation) + graph max-pool + dense.
// MI455X / gfx1250: wave32, WMMA 16x16 tiles. Matmuls use V_WMMA_F32_16X16X4_F32
// (fp32 in/out, matches the fp32 reference). Edge aggregation is the bandwidth
// bottleneck (~1.2 GB of gather+atomic traffic, mostly L2-resident).
// ---------------------------------------------------------------------------

#define N_NODES   100000
#define N_EDGES   3200000
#define F_IN      128
#define H1        32
#define H2        16
#define R_REL     8
#define G_GRAPHS  512

typedef __attribute__((ext_vector_type(2))) float v2f;
typedef __attribute__((ext_vector_type(8))) float v8f;

// ---------------------------- utility kernels ------------------------------

__global__ void fill_zero(float* __restrict__ p, int n) {
  int i = blockIdx.x * blockDim.x + threadIdx.x;
  int stride = gridDim.x * blockDim.x;
  for (; i < n; i += stride) p[i] = 0.0f;
}

// W1[r][i][o] (R,F_IN,H1) + root1[i][o] -> wcat[i][c], c = r*H1+o for c<256,
// root in cols 256..287.  [F_IN x 288]
__global__ void repack_w1(const float* __restrict__ W1,
                          const float* __restrict__ root1,
                          float* __restrict__ wcat) {
  int t = blockIdx.x * blockDim.x + threadIdx.x;
  if (t >= F_IN * 288) return;
  int i = t / 288, c = t % 288;
  float v;
  if (c < R_REL * H1) {
    int r = c >> 5, o = c & 31;
    v = W1[r * F_IN * H1 + i * H1 + o];
  } else {
    v = root1[i * H1 + (c - R_REL * H1)];
  }
  wcat[t] = v;
}

// W2[r][i][o] (R,H1,H2) + root2 -> wcat2[i][c], [H1 x 144]
__global__ void repack_w2(const float* __restrict__ W2,
                          const float* __restrict__ root2,
                          float* __restrict__ wcat) {
  int t = blockIdx.x * blockDim.x + threadIdx.x;
  if (t >= H1 * 144) return;
  int i = t / 144, c = t % 144;
  float v;
  if (c < R_REL * H2) {
    int r = c >> 4, o = c & 15;
    v = W2[r * H1 * H2 + i * H2 + o];
  } else {
    v = root2[i * H2 + (c - R_REL * H2)];
  }
  wcat[t] = v;
}

// ------------------------------- WMMA GEMM ---------------------------------
// Y[N_NODES x cols] = X[N_NODES x K] @ W[K x cols], fp32 via wmma 16x16x4.
// Block = 288 threads = 9 wave32 waves. Block stages one 16-row strip of X in
// LDS (padded stride -> conflict-free); each wave owns one 16x16 output tile.
// N_NODES % 16 == 0, K % 4 == 0, cols % 16 == 0, colTiles % 9 == 0.
__global__ __launch_bounds__(288)
void gemm_wmma(const float* __restrict__ X, const float* __restrict__ W,
               float* __restrict__ Y, int K, int ksh, int cols) {
  extern __shared__ float lds[];              // 16 * (K+4) floats
  const int ldk = K + 4;
  const int rowTile = blockIdx.x;
  const int tid = threadIdx.x;

  // cooperative, coalesced stage of X[rowTile*16 .. +16) into LDS
  for (int idx = tid; idx < (16 << ksh); idx += 288) {
    int r = idx >> ksh, c = idx & (K - 1);
    lds[r * ldk + c] = X[(rowTile * 16 + r) * K + c];
  }
  __syncthreads();

  const int wave = tid >> 5;                  // wave32
  const int lane = tid & 31;
  const int colTile = blockIdx.y * 9 + wave;
  const int colBase = colTile * 16;

  // fp32 16x16x4 fragment layout (ISA 7.12.2):
  //   A: lanes 0-15 hold row M=lane, K=(0,1); lanes 16-31 hold K=(2,3)
  //   B: lanes 0-15 hold col N=lane,  K=(0,1); lanes 16-31 hold K=(2,3)
  const int mrow = lane & 15;
  const int kOff = (lane < 16) ? 0 : 2;
  const int colB = colBase + (lane & 15);

  v8f acc = {};
  for (int k = 0; k < K; k += 4) {
    v2f a, b;
    a.x = lds[mrow * ldk + k + kOff];
    a.y = lds[mrow * ldk + k + kOff + 1];
    b.x = W[(k + kOff) * cols + colB];
    b.y = W[(k + kOff + 1) * cols + colB];
    acc = __builtin_amdgcn_wmma_f32_16x16x4_f32(
        /*neg_a=*/false, a, /*neg_b=*/false, b,
        /*c_mod=*/(short)0, acc, /*reuse_a=*/false, /*reuse_b=*/false);
  }

  // C/D layout: lanes 0-15 rows 0..7, lanes 16-31 rows 8..15
  const int rBase = rowTile * 16 + ((lane < 16) ? 0 : 8);
  const int cOut = colBase + (lane & 15);
#pragma unroll
  for (int j = 0; j < 8; ++j)
    Y[(rBase + j) * cols + cOut] = acc[j];
}

// ----------------------------- edge kernels --------------------------------

__global__ void edge_count(const int* __restrict__ dst,
                           const int* __restrict__ rel,
                           float* __restrict__ cnt, int ne) {
  int e = blockIdx.x * blockDim.x + threadIdx.x;
  if (e >= ne) return;
  atomicAdd(&cnt[dst[e] * R_REL + rel[e]], 1.0f);
}

// one thread per (edge, out-channel); hsh = log2(Hout)
__global__ void edge_agg(const int* __restrict__ src, const int* __restrict__ dst,
                         const int* __restrict__ rel, const float* __restrict__ xw,
                         float* __restrict__ sums, int hsh, int ldx, int total) {
  int gid = blockIdx.x * blockDim.x + threadIdx.x;
  if (gid >= total) return;
  const int Hout = 1 << hsh;
  int e = gid >> hsh;
  int o = gid & (Hout - 1);
  int s = src[e], d = dst[e], r = rel[e];
  float v = xw[s * ldx + r * Hout + o];                 // gather (L2-resident)
  atomicAdd(&sums[d * (R_REL << hsh) + r * Hout + o], v);
}

// h[n][o] = relu( sum_r sums[n,r,o]/max(cnt[n,r],1) + xw_root[n][o] + bias[o] )
__global__ void combine(const float* __restrict__ sums, const float* __restrict__ cnt,
                        const float* __restrict__ xw, const float* __restrict__ bias,
                        float* __restrict__ h, int hsh, int ldx, int total) {
  int t = blockIdx.x * blockDim.x + threadIdx.x;
  if (t >= total) return;
  const int Hout = 1 << hsh;
  int n = t >> hsh;
  int o = t & (Hout - 1);
  float a = xw[n * ldx + (R_REL << hsh) + o] + bias[o]; // root part + bias
#pragma unroll
  for (int r = 0; r < R_REL; ++r) {
    float c = cnt[n * R_REL + r];
    a += sums[n * (R_REL << hsh) + r * Hout + o] / fmaxf(c, 1.0f);
  }
  h[t] = fmaxf(a, 0.0f);
}

// pooled pre-zeroed; h2 >= 0 post-ReLU, so int-bit atomicMax is exact and
// empty graphs stay at 0 (matching the reference's -inf -> 0 fixup).
__global__ void pool_max(const float* __restrict__ h2, const int* __restrict__ batch,
                         float* __restrict__ pooled, int total) {
  int t = blockIdx.x * blockDim.x + threadIdx.x;
  if (t >= total) return;
  int n = t >> 4, o = t & 15;
  float v = h2[t];
  atomicMax((int*)&pooled[batch[n] * H2 + o], __float_as_int(v));
}

__global__ void final_dense(const float* __restrict__ pooled,
                            const float* __restrict__ w,
                            const float* __restrict__ b,
                            float* __restrict__ out) {
  int g = blockIdx.x * blockDim.x + threadIdx.x;
  if (g >= G_GRAPHS) return;
  float a = b[0];
#pragma unroll
  for (int o = 0; o < H2; ++o) a += pooled[g * H2 + o] * w[o];
  out[g] = a;
}

// ------------------------------- launcher ----------------------------------

extern "C" void kernel_launch(void* const* d_in, const int* in_sizes, int n_in,
                              void* d_out, int out_size, void* d_ws, size_t ws_size,
                              hipStream_t stream) {
  const float* x     = (const float*)d_in[0];
  const int*   ei    = (const int*)d_in[1];
  const int*   src   = ei;
  const int*   dst   = ei + N_EDGES;
  const int*   rel   = (const int*)d_in[2];
  const int*   batch = (const int*)d_in[3];
  const float* W1    = (const float*)d_in[4];
  const float* root1 = (const float*)d_in[5];
  const float* b1    = (const float*)d_in[6];
  const float* W2    = (const float*)d_in[7];
  const float* root2 = (const float*)d_in[8];
  const float* b2    = (const float*)d_in[9];
  const float* dw    = (const float*)d_in[10];
  const float* db    = (const float*)d_in[11];
  float* out = (float*)d_out;

  // workspace layout (float units), layer-2 buffers alias the layer-1 xw region
  float* ws     = (float*)d_ws;
  float* wcat1  = ws;                          // 128*288  = 36864
  float* wcat2  = ws + 36864;                  // 32*144   = 4608
  float* cnt    = ws + 41472;                  // N*8      = 800000
  float* h1     = ws + 841472;                 // N*32     = 3200000
  float* h2     = ws + 4041472;                // N*16     = 1600000
  float* pooled = ws + 5641472;                // G*16     = 8192
  float* sums1  = ws + 5649664;                // N*256    = 25600000
  float* big    = ws + 31249664;               // N*288    = 28800000
  float* xw1    = big;                         // [N x 288]
  float* xw2    = big;                         // [N x 144] (after layer 1 done)
  float* sums2  = big + 14400000;              // [N x 128]

  // init + weight repack + edge counts (counts shared by both layers)
  fill_zero<<<2048, 256, 0, stream>>>(sums1, N_NODES * 256);
  fill_zero<<<512, 256, 0, stream>>>(cnt, N_NODES * R_REL);
  fill_zero<<<32, 256, 0, stream>>>(pooled, G_GRAPHS * H2);
  repack_w1<<<(F_IN * 288 + 255) / 256, 256, 0, stream>>>(W1, root1, wcat1);
  repack_w2<<<(H1 * 144 + 255) / 256, 256, 0, stream>>>(W2, root2, wcat2);
  edge_count<<<(N_EDGES + 255) / 256, 256, 0, stream>>>(dst, rel, cnt, N_EDGES);

  // ---- layer 1 ----
  gemm_wmma<<<dim3(N_NODES / 16, 2), 288, 16 * (F_IN + 4) * sizeof(float), stream>>>(
      x, wcat1, xw1, F_IN, 7, 288);
  edge_agg<<<(N_EDGES * 32 + 255) / 256, 256, 0, stream>>>(
      src, dst, rel, xw1, sums1, 5, 288, N_EDGES * 32);
  combine<<<(N_NODES * 32 + 255) / 256, 256, 0, stream>>>(
      sums1, cnt, xw1, b1, h1, 5, 288, N_NODES * 32);

  // ---- layer 2 ----
  fill_zero<<<2048, 256, 0, stream>>>(sums2, N_NODES * 128);
  gemm_wmma<<<dim3(N_NODES / 16, 1), 288, 16 * (H1 + 4) * sizeof(float), stream>>>(
      h1, wcat2, xw2, H1, 5, 144);
  edge_agg<<<(N_EDGES * 16 + 255) / 256, 256, 0, stream>>>(
      src, dst, rel, xw2, sums2, 4, 144, N_EDGES * 16);
  combine<<<(N_NODES * 16 + 255) / 256, 256, 0, stream>>>(
      sums2, cnt, xw2, b2, h2, 4, 144, N_NODES * 16);

  // ---- pool + dense ----
  pool_max<<<(N_NODES * 16 + 255) / 256, 256, 0, stream>>>(h2, batch, pooled, N_NODES * 16);
  final_dense<<<2, 256, 0, stream>>>(pooled, dw, db, out);
}